// ChebGNNEncoder_26706106646650
// MI455X (gfx1250) — compile-verified
//
#include <hip/hip_runtime.h>
#include <hip/hip_bf16.h>

typedef float v2f __attribute__((ext_vector_type(2)));
typedef float v8f __attribute__((ext_vector_type(8)));

#define N_NODES 20000
#define N_EDGES 320000
#define F_IN    350
#define F_PAD0  352   // F_IN padded to multiple of 32

struct Pages { const float* p[5]; };

// ---------------- utility kernels ----------------

__global__ void zero_f(float* __restrict__ dst, int count) {
    int i = blockIdx.x * 256 + threadIdx.x;
    if (i < count) dst[i] = 0.0f;
}

__global__ void pad_x(float* __restrict__ xp, const float* __restrict__ x, int count) {
    int i = blockIdx.x * 256 + threadIdx.x;
    if (i >= count) return;
    int n = i / F_PAD0;
    int f = i - n * F_PAD0;
    xp[i] = (f < F_IN) ? x[(size_t)n * F_IN + f] : 0.0f;
}

__global__ void deg_kernel(const int* __restrict__ row, const int* __restrict__ col,
                           float* __restrict__ deg, int ne) {
    int e = blockIdx.x * 256 + threadIdx.x;
    if (e >= ne) return;
    int r = row[e], c = col[e];
    if (r != c) atomicAdd(&deg[r], 1.0f);
}

__global__ void dinv_kernel(const float* __restrict__ deg, float* __restrict__ dinv, int n) {
    int i = blockIdx.x * 256 + threadIdx.x;
    if (i >= n) return;
    float d = deg[i];
    dinv[i] = (d > 0.0f) ? rsqrtf(fmaxf(d, 1e-12f)) : 0.0f;
}

__global__ void wedge_kernel(const int* __restrict__ row, const int* __restrict__ col,
                             const float* __restrict__ dinv, float* __restrict__ w, int ne) {
    int e = blockIdx.x * 256 + threadIdx.x;
    if (e >= ne) return;
    int r = row[e], c = col[e];
    w[e] = (r != c) ? (-dinv[r] * dinv[c]) : 0.0f;
}

// Pack weights: Wp[(k*Fp + i)*Fout + o] = Wc[k][i][o] (+ Wl[i][o] when k==0), zero pad rows
__global__ void pack_w(float* __restrict__ Wp, const float* __restrict__ Wc,
                       const float* __restrict__ Wl, int Fin, int Fp, int Fout, int total) {
    int i = blockIdx.x * 256 + threadIdx.x;
    if (i >= total) return;
    int o  = i % Fout;
    int kt = i / Fout;
    int k  = kt / Fp;
    int fi = kt - k * Fp;
    float v = 0.0f;
    if (fi < Fin) {
        v = Wc[((size_t)k * Fin + fi) * Fout + o];
        if (k == 0) v += Wl[(size_t)fi * Fout + o];
    }
    Wp[i] = v;
}

__global__ void pack_b(float* __restrict__ be, const float* __restrict__ bc,
                       const float* __restrict__ bl, int Fout) {
    int i = blockIdx.x * 256 + threadIdx.x;
    if (i < Fout) be[i] = bc[i] + bl[i];
}

// ---------------- SpMM propagation ----------------

// Tdst = (prev ? -prev : 0)
__global__ void prop_init(float* __restrict__ dst, const float* __restrict__ prev, int count) {
    int i = blockIdx.x * 256 + threadIdx.x;
    if (i >= count) return;
    dst[i] = prev ? -prev[i] : 0.0f;
}

// Tdst[col[e]] += scale * w[e] * Tsrc[row[e]]   (one wave per edge, lanes sweep features)
__global__ __launch_bounds__(256)
void prop_edge(const int* __restrict__ row, const int* __restrict__ col,
               const float* __restrict__ w, const float* __restrict__ Tsrc,
               float* __restrict__ Tdst, int Fp, float scale, int ne) {
    int e = blockIdx.x * 8 + (threadIdx.x >> 5);
    if (e >= ne) return;
    int lane = threadIdx.x & 31;
    float wv = w[e] * scale;
    if (wv == 0.0f) return;
    const float* src = Tsrc + (size_t)row[e] * Fp;
    float*       dst = Tdst + (size_t)col[e] * Fp;
    for (int f = lane * 4; f < Fp; f += 128) {
        float4 v = *(const float4*)(src + f);
        atomicAdd(dst + f + 0, wv * v.x);
        atomicAdd(dst + f + 1, wv * v.y);
        atomicAdd(dst + f + 2, wv * v.z);
        atomicAdd(dst + f + 3, wv * v.w);
    }
}

// ---------------- fused Chebyshev GEMM (fp32 WMMA) ----------------
// out[N x Fout] = relu( sum_k Tk @ Wp_k + bias ), K-dim = 5*Fp, LDS-staged tiles.
// Block: 256 thr = 8 waves (2 M x 4 N), tile 32(M) x 64(N), K-chunk 32.

#define AS_LD 36
#define BS_LD 68

__global__ __launch_bounds__(256)
void cheb_gemm(Pages pg, const float* __restrict__ Wp, const float* __restrict__ bias,
               float* __restrict__ out, int Fp, int Fout, int Ktot) {
    __shared__ float As[32 * AS_LD];
    __shared__ float Bs[32 * BS_LD];

    const int tid   = threadIdx.x;
    const int lane  = tid & 31;
    const int wid   = tid >> 5;
    const int wm    = (wid & 1) * 16;     // wave M offset inside block tile
    const int wn    = (wid >> 1) * 16;    // wave N offset inside block tile
    const int mlane = lane & 15;          // == nlane
    const int khalf = (lane >> 4) * 2;    // 0 or 2

    const int m0 = blockIdx.x * 32;
    const int n0 = blockIdx.y * 64;

    // A-tile loader indices: 32 rows x 8 float4
    const int ar = tid >> 3;
    const int ac = (tid & 7) * 4;
    // B-tile loader indices: 16 rows x 16 float4, two passes
    const int br = tid >> 4;
    const int bc4 = (tid & 15) * 4;

    v8f acc = {};

    for (int kt0 = 0; kt0 < Ktot; kt0 += 32) {
        const int page = kt0 / Fp;              // chunk never straddles pages (Fp % 32 == 0)
        const int i0   = kt0 - page * Fp;
        const float* __restrict__ Ag = pg.p[page];

        // stage A: rows m0..m0+31, cols i0..i0+31
        {
            float4 v = *(const float4*)(Ag + (size_t)(m0 + ar) * Fp + i0 + ac);
            *(float4*)(&As[ar * AS_LD + ac]) = v;
        }
        // stage B: Wp rows kt0..kt0+31, cols n0..n0+63
        {
            float4 v0 = *(const float4*)(Wp + (size_t)(kt0 + br) * Fout + n0 + bc4);
            *(float4*)(&Bs[br * BS_LD + bc4]) = v0;
            float4 v1 = *(const float4*)(Wp + (size_t)(kt0 + br + 16) * Fout + n0 + bc4);
            *(float4*)(&Bs[(br + 16) * BS_LD + bc4]) = v1;
        }
        __syncthreads();

#pragma unroll
        for (int kk = 0; kk < 32; kk += 4) {
            v2f a, b;
            const float* ap = &As[(wm + mlane) * AS_LD + kk + khalf];
            a.x = ap[0];
            a.y = ap[1];
            const float* bp = &Bs[(kk + khalf) * BS_LD + wn + mlane];
            b.x = bp[0];
            b.y = bp[BS_LD];
            acc = __builtin_amdgcn_wmma_f32_16x16x4_f32(false, a, false, b,
                                                        (short)0, acc, false, false);
        }
        __syncthreads();
    }

    // epilogue: D layout — VGPR r: lanes 0-15 -> M=r, lanes 16-31 -> M=r+8
    const int colg  = n0 + wn + mlane;
    const int mbase = (lane >> 4) * 8;
    const float bv  = bias[colg];
#pragma unroll
    for (int r = 0; r < 8; ++r) {
        int rowg = m0 + wm + mbase + r;
        float v = acc[r] + bv;
        out[(size_t)rowg * Fout + colg] = fmaxf(v, 0.0f);
    }
}

// ---------------- host orchestration ----------------

extern "C" void kernel_launch(void* const* d_in, const int* in_sizes, int n_in,
                              void* d_out, int out_size, void* d_ws, size_t ws_size,
                              hipStream_t stream) {
    const int Nn = N_NODES, Ne = N_EDGES;

    const float* x   = (const float*)d_in[0];
    const int*   ei  = (const int*)d_in[1];
    const float* Wc[3] = {(const float*)d_in[2], (const float*)d_in[6],  (const float*)d_in[10]};
    const float* bcv[3] = {(const float*)d_in[3], (const float*)d_in[7],  (const float*)d_in[11]};
    const float* Wl[3] = {(const float*)d_in[4], (const float*)d_in[8],  (const float*)d_in[12]};
    const float* blv[3] = {(const float*)d_in[5], (const float*)d_in[9],  (const float*)d_in[13]};

    const int* rowp = ei;
    const int* colp = ei + Ne;

    // workspace carve-up (floats)
    float* ws = (float*)d_ws;
    const size_t RSZ = (size_t)Nn * F_PAD0;
    float* R[6];
    for (int r = 0; r < 6; ++r) R[r] = ws + (size_t)r * RSZ;
    float* Wpk  = ws + 6 * RSZ;                 // 5*352*256 max
    float* be   = Wpk + (size_t)5 * F_PAD0 * 256;
    float* wE   = be + 256;
    float* degb = wE + Ne;
    float* dnv  = degb + Nn;

    // edge normalization
    zero_f<<<(Nn + 255) / 256, 256, 0, stream>>>(degb, Nn);
    deg_kernel<<<(Ne + 255) / 256, 256, 0, stream>>>(rowp, colp, degb, Ne);
    dinv_kernel<<<(Nn + 255) / 256, 256, 0, stream>>>(degb, dnv, Nn);
    wedge_kernel<<<(Ne + 255) / 256, 256, 0, stream>>>(rowp, colp, dnv, wE, Ne);

    // pad input features into R0
    pad_x<<<((int)RSZ + 255) / 256, 256, 0, stream>>>(R[0], x, (int)RSZ);

    auto run_layer = [&](const float* T0, float* T1, float* T2, float* T3, float* T4,
                         const float* Wc_, const float* bc_, const float* Wl_, const float* bl_,
                         int Fin, int Fp, int Fout, float* outp) {
        int wcnt = 5 * Fp * Fout;
        pack_w<<<(wcnt + 255) / 256, 256, 0, stream>>>(Wpk, Wc_, Wl_, Fin, Fp, Fout, wcnt);
        pack_b<<<(Fout + 255) / 256, 256, 0, stream>>>(be, bc_, bl_, Fout);

        int cnt = Nn * Fp;
        dim3 gi((cnt + 255) / 256);
        dim3 ge(Ne / 8);
        prop_init<<<gi, 256, 0, stream>>>(T1, nullptr, cnt);
        prop_edge<<<ge, 256, 0, stream>>>(rowp, colp, wE, T0, T1, Fp, 1.0f, Ne);
        prop_init<<<gi, 256, 0, stream>>>(T2, T0, cnt);
        prop_edge<<<ge, 256, 0, stream>>>(rowp, colp, wE, T1, T2, Fp, 2.0f, Ne);
        prop_init<<<gi, 256, 0, stream>>>(T3, T1, cnt);
        prop_edge<<<ge, 256, 0, stream>>>(rowp, colp, wE, T2, T3, Fp, 2.0f, Ne);
        prop_init<<<gi, 256, 0, stream>>>(T4, T2, cnt);
        prop_edge<<<ge, 256, 0, stream>>>(rowp, colp, wE, T3, T4, Fp, 2.0f, Ne);

        Pages pg;
        pg.p[0] = T0; pg.p[1] = T1; pg.p[2] = T2; pg.p[3] = T3; pg.p[4] = T4;
        dim3 gg(Nn / 32, Fout / 64);
        cheb_gemm<<<gg, 256, 0, stream>>>(pg, Wpk, be, outp, Fp, Fout, 5 * Fp);
    };

    // layer 0: 350(pad 352) -> 256, output to R5
    run_layer(R[0], R[1], R[2], R[3], R[4], Wc[0], bcv[0], Wl[0], blv[0],
              F_IN, F_PAD0, 256, R[5]);
    // layer 1: 256 -> 256, T0 = R5, output to R0
    run_layer(R[5], R[1], R[2], R[3], R[4], Wc[1], bcv[1], Wl[1], blv[1],
              256, 256, 256, R[0]);
    // layer 2: 256 -> 64, T0 = R0, output to d_out
    run_layer(R[0], R[1], R[2], R[3], R[4], Wc[2], bcv[2], Wl[2], blv[2],
              256, 256, 64, (float*)d_out);
}